// InferenceDynamicEmbeddingCollection_70265664962867
// MI455X (gfx1250) — compile-verified
//
#include <hip/hip_runtime.h>
#include <hip/hip_bf16.h>

// ---- problem constants (fixed by the reference) ----
#define FEATS     8
#define BL        81920            // B*L = 4096*20
#define VOCAB_N   1000000u
#define DIM       128              // floats per row
#define ROW_BYTES (DIM * 4)        // 512 B = wave32 x 16B
#define ROWS      (FEATS * BL)     // 655360 output rows
#define RPW       8                // rows per wave
#define WPB       8                // waves per block (256 threads)

typedef float v4f __attribute__((ext_vector_type(4)));
typedef int   v4i __attribute__((ext_vector_type(4)));

#if __has_builtin(__builtin_amdgcn_global_load_async_to_lds_b128) && \
    __has_builtin(__builtin_amdgcn_global_store_async_from_lds_b128)
#define USE_ASYNC_LDS 1
#else
#define USE_ASYNC_LDS 0
#endif

#if USE_ASYNC_LDS
// -------- CDNA5 async global<->LDS DMA path (ASYNCcnt) --------
// builtin params are strictly typed: (AS1 v4i*, AS3 v4i*, imm offset, imm cpol)
typedef __attribute__((address_space(1))) v4i gv4i;   // global ("__device__") int4
typedef __attribute__((address_space(3))) v4i lv4i;   // LDS ("__shared__") int4

__device__ __forceinline__ void wait_async0() {
#if __has_builtin(__builtin_amdgcn_s_wait_asynccnt)
    __builtin_amdgcn_s_wait_asynccnt(0);
#else
    asm volatile("s_wait_asynccnt 0" ::: "memory");
#endif
}

__global__ void __launch_bounds__(256)
emb_gather_async(const int* __restrict__ values,
                 const float* __restrict__ table,
                 float* __restrict__ out) {
    __shared__ char lds[WPB * RPW * ROW_BYTES];   // 32 KB / block
    const unsigned wv   = threadIdx.x >> 5;       // wave in block
    const unsigned lane = threadIdx.x & 31u;
    const unsigned gw   = blockIdx.x * WPB + wv;  // global wave index
    const unsigned r0   = gw * RPW;               // 8 consecutive rows

    // 8-row group never straddles a feature boundary (BL % 8 == 0)
    const unsigned f    = r0 / BL;
    const unsigned pos0 = r0 - f * BL;
    const unsigned srcf = f ^ 4u;   // perm = [4,5,6,7,0,1,2,3]

    // two coalesced b128 loads fetch all 8 ids for this wave
    const v4i* idp = reinterpret_cast<const v4i*>(values + (size_t)srcf * BL + pos0);
    const v4i ia = idp[0];
    const v4i ib = idp[1];
    unsigned slot[RPW] = {
        (unsigned)ia.x % VOCAB_N, (unsigned)ia.y % VOCAB_N,
        (unsigned)ia.z % VOCAB_N, (unsigned)ia.w % VOCAB_N,
        (unsigned)ib.x % VOCAB_N, (unsigned)ib.y % VOCAB_N,
        (unsigned)ib.z % VOCAB_N, (unsigned)ib.w % VOCAB_N };

    char* lbase = lds + (size_t)wv * (RPW * ROW_BYTES) + (size_t)lane * 16u;

    // DMA gather: table row -> LDS (8 in flight, no VGPR data movement)
#pragma unroll
    for (int i = 0; i < RPW; ++i) {
        const char* g = (const char*)table + (size_t)slot[i] * ROW_BYTES + lane * 16u;
        __builtin_amdgcn_global_load_async_to_lds_b128(
            (gv4i*)g, (lv4i*)(lbase + i * ROW_BYTES), 0, 0);
    }
    wait_async0();   // all 8 rows resident in LDS

    // DMA stream: LDS -> out, TH=NT (keep L2 for table-row reuse)
#pragma unroll
    for (int i = 0; i < RPW; ++i) {
        char* g = (char*)out + (size_t)(r0 + i) * ROW_BYTES + lane * 16u;
        __builtin_amdgcn_global_store_async_from_lds_b128(
            (gv4i*)g, (lv4i*)(lbase + i * ROW_BYTES), 0, 1 /*TH_STORE_NT*/);
    }
    // S_ENDPGM waits all counters -> stores complete before kernel end.
}

#else
// -------- fallback: proven VGPR b128 gather path --------
__global__ void __launch_bounds__(256)
emb_gather_vgpr(const int* __restrict__ values,
                const float* __restrict__ table,
                float* __restrict__ out) {
    const unsigned tid  = blockIdx.x * blockDim.x + threadIdx.x;
    const unsigned w    = tid >> 5;
    const unsigned lane = tid & 31u;
    const unsigned r0   = w * 4u;
    if (r0 >= ROWS) return;

    const unsigned f    = r0 / BL;
    const unsigned pos0 = r0 - f * BL;
    const unsigned srcf = f ^ 4u;

    const v4i ids = *reinterpret_cast<const v4i*>(values + (size_t)srcf * BL + pos0);
    unsigned slot[4] = { (unsigned)ids.x % VOCAB_N, (unsigned)ids.y % VOCAB_N,
                         (unsigned)ids.z % VOCAB_N, (unsigned)ids.w % VOCAB_N };
    v4f v[4];
#pragma unroll
    for (int i = 0; i < 4; ++i)
        v[i] = reinterpret_cast<const v4f*>(table + (size_t)slot[i] * DIM)[lane];
#pragma unroll
    for (int i = 0; i < 4; ++i)
        __builtin_nontemporal_store(
            v[i], reinterpret_cast<v4f*>(out + (size_t)(r0 + i) * DIM) + lane);
}
#endif

extern "C" void kernel_launch(void* const* d_in, const int* in_sizes, int n_in,
                              void* d_out, int out_size, void* d_ws, size_t ws_size,
                              hipStream_t stream) {
    (void)in_sizes; (void)n_in; (void)out_size; (void)d_ws; (void)ws_size;
    const int*   values = (const int*)d_in[0];    // [F, B*L] int32
    const float* table  = (const float*)d_in[1];  // [VOCAB, DIM] f32
    float*       out    = (float*)d_out;          // [F, B*L, DIM] f32

#if USE_ASYNC_LDS
    const unsigned waves  = ROWS / RPW;           // 81920
    const unsigned blocks = waves / WPB;          // 10240
    emb_gather_async<<<dim3(blocks), dim3(256), 0, stream>>>(values, table, out);
#else
    const unsigned waves  = ROWS / 4u;            // 163840
    const unsigned blocks = (waves * 32u) / 256u; // 20480
    emb_gather_vgpr<<<dim3(blocks), dim3(256), 0, stream>>>(values, table, out);
#endif
}